// LossRetina3_18339510354528
// MI455X (gfx1250) — compile-verified
//
#include <hip/hip_runtime.h>
#include <math.h>

// ---------------------------------------------------------------------------
// Retina loss (GFL pos/neg + DFL + box) for B=32, N=10842.
// HBM-bound: ~60 MB in, 1 float out -> ~2.6 us floor at 23.3 TB/s.
// Pass 1: one thread per anchor, b128 streaming loads, per-wave reduction via
//         V_WMMA_F32_16X16X4_F32 (A = lane partials, B = ones -> row sums),
//         cross-wave via LDS, per-block 5-float partial to workspace.
// Pass 2: one wave; lane b folds its row's partials, applies weights and
//         1/clip(num_pos, 2^-10), final 32-lane WMMA reduction, /B, store.
// Fully deterministic (fixed reduction trees, no float atomics).
// ---------------------------------------------------------------------------

typedef __attribute__((ext_vector_type(2))) float v2f;
typedef __attribute__((ext_vector_type(8))) float v8f;

#define LOSS_B 32
#define LOSS_REGS 8          // REG_MAX + 1
#define LOSS_NCLS 3
#define LOSS_EPS 0.0009765625f

// Sum a float across all 32 lanes of the wave using one f32 WMMA.
// A layout (16x4 f32): lanes 0-15 VGPR0 = A[m,0], lanes 16-31 VGPR0 = A[m,2].
// With a.y = 0 and B = ones: D[m,n] = v[m] + v[m+16] for every n.
// D layout: lane L<16 holds rows 0..7 in its 8 regs, lane L>=16 rows 8..15,
// so a lane-local sum + shfl_xor(16) gives the full 32-lane total on every lane.
__device__ __forceinline__ float wave_sum_wmma(float v) {
    v2f a; a.x = v;    a.y = 0.0f;
    v2f b; b.x = 1.0f; b.y = 1.0f;
    v8f c = {};
    v8f d = __builtin_amdgcn_wmma_f32_16x16x4_f32(
        /*neg_a=*/false, a, /*neg_b=*/false, b,
        /*c_mod=*/(short)0, c, /*reuse_a=*/false, /*reuse_b=*/false);
    float s = ((d[0] + d[1]) + (d[2] + d[3])) + ((d[4] + d[5]) + (d[6] + d[7]));
    s += __shfl_xor(s, 16, 32);
    return s;
}

__device__ __forceinline__ float sigmoidf_fast(float x) {
    return 1.0f / (1.0f + __expf(-x));
}

__global__ __launch_bounds__(256)
void loss_partial_kernel(const float* __restrict__ preg,
                         const float* __restrict__ pcls,
                         const float* __restrict__ gre,
                         float* __restrict__ partial,  // [B * gridDim.x][5]
                         int N) {
    const int b = blockIdx.y;
    const int n = blockIdx.x * blockDim.x + threadIdx.x;
    const bool valid = (n < N);
    // Clamp to a safe in-row address for tail lanes; their contribution is zeroed.
    const size_t idx = (size_t)b * N + (valid ? n : 0);

    // ---- streamed loads (b128 where possible) ----
    const float4* pr4 = (const float4*)(preg + idx * 32);
    float4 r[8];
#pragma unroll
    for (int i = 0; i < 8; ++i) r[i] = pr4[i];

    const float4* g4 = (const float4*)(gre + idx * 8);
    float4 g0 = g4[0];   // iou, gcls0, gcls1, gcls2
    float4 g1 = g4[1];   // gt_ltrb[0..3]

    const float* pc = pcls + idx * 3;
    float c0 = pc[0], c1 = pc[1], c2 = pc[2];

    // ---- per-anchor math (all fp32, matches reference) ----
    const float iou  = g0.x;
    const float mask = (iou > 0.0f) ? 1.0f : 0.0f;

    float ps[LOSS_NCLS] = { sigmoidf_fast(c0), sigmoidf_fast(c1), sigmoidf_fast(c2) };
    const float wcls = fmaxf(ps[0], fmaxf(ps[1], ps[2]));

    float acc_cnt = mask;
    float acc_box = (1.0f - iou) * wcls * mask;

    // DFL: 4 sides x log_softmax over 8 bins; gather at tl, tl+1.
    float lg[32];
#pragma unroll
    for (int i = 0; i < 8; ++i) {
        lg[4 * i + 0] = r[i].x; lg[4 * i + 1] = r[i].y;
        lg[4 * i + 2] = r[i].z; lg[4 * i + 3] = r[i].w;
    }
    const float gt[4] = { g1.x, g1.y, g1.z, g1.w };
    float acc_dfl = 0.0f;
#pragma unroll
    for (int dside = 0; dside < 4; ++dside) {
        float t  = fminf(fmaxf(gt[dside], 0.0f), 6.999f);
        int   tl = (int)t;
        float wr = t - (float)tl;

        float m = lg[dside * 8];
#pragma unroll
        for (int j = 1; j < 8; ++j) m = fmaxf(m, lg[dside * 8 + j]);

        float esum = 0.0f, ltl = 0.0f, ltr = 0.0f;
#pragma unroll
        for (int j = 0; j < 8; ++j) {
            float lj = lg[dside * 8 + j];
            esum += __expf(lj - m);
            ltl = (j == tl)     ? lj : ltl;
            ltr = (j == tl + 1) ? lj : ltr;
        }
        float lse = m + __logf(esum);
        acc_dfl += -((ltl - lse) * (1.0f - wr) + (ltr - lse) * wr);
    }
    acc_dfl *= mask;

    // GFL pos/neg over 3 classes.
    const float gc[LOSS_NCLS] = { g0.y, g0.z, g0.w };
    float acc_gp = 0.0f, acc_gn = 0.0f;
#pragma unroll
    for (int cc = 0; cc < LOSS_NCLS; ++cc) {
        float p     = fminf(fmaxf(ps[cc], 1e-6f), 1.0f - 1e-6f);
        float score = gc[cc] * iou;
        float lp    = __logf(p);
        float l1mp  = log1pf(-p);
        float bce   = -(score * lp + (1.0f - score) * l1mp);
        float diff  = fabsf(score - p);
        acc_gp += diff * diff * bce;
        acc_gn += p * p * (-l1mp);
    }
    acc_gp *= mask;
    acc_gn *= (1.0f - mask);

    if (!valid) { acc_cnt = acc_box = acc_dfl = acc_gp = acc_gn = 0.0f; }

    // ---- wave reduction via WMMA (EXEC all-ones: every lane participates) ----
    float w0 = wave_sum_wmma(acc_cnt);
    float w1 = wave_sum_wmma(acc_box);
    float w2 = wave_sum_wmma(acc_dfl);
    float w3 = wave_sum_wmma(acc_gp);
    float w4 = wave_sum_wmma(acc_gn);

    // ---- cross-wave reduction via LDS ----
    __shared__ float lds[8][5];
    const int wave = threadIdx.x >> 5;
    const int lane = threadIdx.x & 31;
    if (lane == 0) {
        lds[wave][0] = w0; lds[wave][1] = w1; lds[wave][2] = w2;
        lds[wave][3] = w3; lds[wave][4] = w4;
    }
    __syncthreads();
    if (threadIdx.x < 5) {
        float s = 0.0f;
        const int nw = blockDim.x >> 5;
        for (int w = 0; w < nw; ++w) s += lds[w][threadIdx.x];
        partial[((size_t)b * gridDim.x + blockIdx.x) * 5 + threadIdx.x] = s;
    }
}

__global__ void loss_final_kernel(const float* __restrict__ partial,
                                  float* __restrict__ out,
                                  int nblk) {
    // One wave32; lane index = batch row (B == 32).
    const int b = threadIdx.x;
    float cnt = 0.0f, box = 0.0f, dfl = 0.0f, gp = 0.0f, gn = 0.0f;
    for (int j = 0; j < nblk; ++j) {
        const float* p = partial + ((size_t)b * nblk + j) * 5;
        cnt += p[0]; box += p[1]; dfl += p[2]; gp += p[3]; gn += p[4];
    }
    float np  = fmaxf(cnt, LOSS_EPS);
    float row = (2.0f * box + 0.25f * dfl + gp + gn) / np;
    float total = wave_sum_wmma(row) * (1.0f / (float)LOSS_B);
    if (b == 0) out[0] = total;
}

extern "C" void kernel_launch(void* const* d_in, const int* in_sizes, int n_in,
                              void* d_out, int out_size, void* d_ws, size_t ws_size,
                              hipStream_t stream) {
    const float* preg = (const float*)d_in[0];  // (B, N, 32) f32
    const float* pcls = (const float*)d_in[1];  // (B, N, 3)  f32
    const float* gre  = (const float*)d_in[2];  // (B, N, 8)  f32
    float* out = (float*)d_out;
    float* ws  = (float*)d_ws;                  // B * nblk_x * 5 floats

    const int N = in_sizes[0] / (LOSS_B * 32);  // 10842
    const int nblk_x = (N + 255) / 256;         // 43

    dim3 grid(nblk_x, LOSS_B);
    loss_partial_kernel<<<grid, 256, 0, stream>>>(preg, pcls, gre, ws, N);
    loss_final_kernel<<<1, 32, 0, stream>>>(ws, out, nblk_x);
}